// MAGLtopkv6_28819230556615
// MI455X (gfx1250) — compile-verified
//
#include <hip/hip_runtime.h>
#include <hip/hip_bf16.h>

// Problem constants (from reference setup)
#define Qn    4096
#define NCH   32
#define Cn    256
#define BATCH 8
#define Mn    12288      // (CAV-1)*Q
#define TOPKN 1024

typedef __attribute__((ext_vector_type(2))) float v2f;
typedef __attribute__((ext_vector_type(8))) float v8f;

// ---------------- K0: fused passthrough copy + confidence keys ----------------
// one wave per (q,n) row of 256 floats: copy to out, dot with heat_w, emit key
__global__ void k_main(const float* __restrict__ x, const float* __restrict__ hw,
                       const float* __restrict__ hb, float* __restrict__ out,
                       unsigned* __restrict__ keys) {
    int lane = threadIdx.x & 31;
    int r = blockIdx.x * 8 + (threadIdx.x >> 5);   // [0, Q*N): r = q*32 + n
    int q = r >> 5, n = r & 31;
    const float4* p = (const float4*)(x + (size_t)r * Cn + lane * 8);
    const float4* h = (const float4*)(hw + lane * 8);
    float4 a0 = p[0], a1 = p[1];
    float4 h0 = h[0], h1 = h[1];
    float4* o = (float4*)(out + (size_t)r * Cn + lane * 8);
    o[0] = a0; o[1] = a1;
    float s = a0.x*h0.x + a0.y*h0.y + a0.z*h0.z + a0.w*h0.w
            + a1.x*h1.x + a1.y*h1.y + a1.z*h1.z + a1.w*h1.w;
    for (int off = 16; off; off >>= 1) s += __shfl_xor(s, off, 32);
    if (lane == 0 && (n & 3) != 0) {
        int b = n >> 2, cc = n & 3;
        int m = (cc - 1) * Qn + q;
        float z = s + hb[0];                       // sigmoid monotonic -> skip
        unsigned bits = __float_as_uint(z);
        unsigned key = bits ^ (((int)bits < 0) ? 0xFFFFFFFFu : 0x80000000u);
        keys[b * Mn + m] = key;
    }
}

// ---------------- K1: 4x4 transform of reference points ----------------
__global__ void k_xyz(const float* __restrict__ ref, const float* __restrict__ pt,
                      float* __restrict__ X, float* __restrict__ Y, float* __restrict__ Z) {
    int t = blockIdx.x * 256 + threadIdx.x;   // N*Q threads
    int n = t >> 12, q = t & 4095;
    const float* rp = ref + (size_t)(n * Qn + q) * 3;
    float rx = rp[0] * 102.4f - 51.2f;
    float ry = rp[1] * 102.4f - 51.2f;
    float rz = rp[2] * 8.0f   - 5.0f;
    const float* tm = pt + n * 64;            // pairwise_t_matrix[b][c][0]
    X[t] = tm[0]*rx + tm[1]*ry + tm[2]*rz  + tm[3];
    Y[t] = tm[4]*rx + tm[5]*ry + tm[6]*rz  + tm[7];
    Z[t] = tm[8]*rx + tm[9]*ry + tm[10]*rz + tm[11];
}

// ---------------- K2: exact top-1024 set via 4-pass radix threshold select ----------------
__global__ void k_select(const unsigned* __restrict__ keys, int* __restrict__ topidx) {
    __shared__ unsigned hist[256];
    __shared__ unsigned s_sel, s_need, s_cntG, s_cntE;
    int b = blockIdx.x, tid = threadIdx.x;
    const unsigned* kb = keys + b * Mn;
    unsigned prefix = 0, prefmask = 0, need = TOPKN;
    for (int by = 3; by >= 0; --by) {
        hist[tid] = 0;
        __syncthreads();
        for (int m = tid; m < Mn; m += 256) {
            unsigned k = kb[m];
            if ((k & prefmask) == prefix) atomicAdd(&hist[(k >> (8 * by)) & 255], 1u);
        }
        __syncthreads();
        if (tid == 0) {
            unsigned cum = 0, nn = need; int sel = 0;
            for (int bin = 255; bin >= 0; --bin) {
                unsigned c = hist[bin];
                if (cum + c >= need) { sel = bin; nn = need - cum; break; }
                cum += c;
            }
            s_sel = (unsigned)sel; s_need = nn;
        }
        __syncthreads();
        prefix  |= s_sel << (8 * by);
        prefmask |= 0xFFu << (8 * by);
        need = s_need;
        __syncthreads();
    }
    unsigned T = prefix;
    if (tid == 0) { s_cntG = 0; s_cntE = 0; }
    __syncthreads();
    unsigned nGT = TOPKN - need;               // count strictly greater than T
    int* tb = topidx + b * TOPKN;
    for (int m = tid; m < Mn; m += 256) {
        unsigned k = kb[m];
        if (k > T) { unsigned p = atomicAdd(&s_cntG, 1u); tb[p] = m; }
        else if (k == T) { unsigned p = atomicAdd(&s_cntE, 1u); if (p < need) tb[nGT + p] = m; }
    }
}

// ---------------- K3: gather top features + packed WMMA B-operands ----------------
__global__ void k_gather(const float* __restrict__ x, const int* __restrict__ topidx,
                         const float* __restrict__ X, const float* __restrict__ Y,
                         const float* __restrict__ Z, float* __restrict__ tf,
                         float2* __restrict__ bpk0, float2* __restrict__ bpk1) {
    int lane = threadIdx.x & 31;
    int w = blockIdx.x * 8 + (threadIdx.x >> 5);   // [0, B*TOPK)
    int b = w >> 10, k = w & 1023;
    int m = topidx[b * TOPKN + k];
    int cc = 1 + (m >> 12), q = m & 4095, n = b * 4 + cc;
    const float4* src = (const float4*)(x + (size_t)(q * NCH + n) * Cn + lane * 8);
    float4* dst = (float4*)(tf + (size_t)(b * TOPKN + k) * Cn + lane * 8);
    dst[0] = src[0]; dst[1] = src[1];
    if (lane == 0) {
        int idx = n * Qn + q;
        float px = X[idx], py = Y[idx], pz = Z[idx];
        int o = b * TOPKN + k;
        bpk0[o] = make_float2(-2.0f * px, -2.0f * py);                 // K=0,1
        bpk1[o] = make_float2(-2.0f * pz, px*px + py*py + pz*pz);      // K=2,3
    }
}

// branchless insertion into sorted ascending 5-list (pure v_cndmask)
__device__ __forceinline__ void ins5(float dd, int id,
    float& d0, float& d1, float& d2, float& d3, float& d4,
    int& i0, int& i1, int& i2, int& i3, int& i4) {
    bool c0 = dd < d0, c1 = dd < d1, c2 = dd < d2, c3 = dd < d3, c4 = dd < d4;
    d4 = c3 ? d3 : (c4 ? dd : d4);  i4 = c3 ? i3 : (c4 ? id : i4);
    d3 = c2 ? d2 : (c3 ? dd : d3);  i3 = c2 ? i2 : (c3 ? id : i3);
    d2 = c1 ? d1 : (c2 ? dd : d2);  i2 = c1 ? i1 : (c2 ? id : i2);
    d1 = c0 ? d0 : (c1 ? dd : d1);  i1 = c0 ? i0 : (c1 ? id : i1);
    d0 = c0 ? dd : d0;              i0 = c0 ? id : i0;
}

// ---------------- K4: WMMA distances + 5-NN + feature max + ego write ----------------
// one wave per (batch, 16-query tile); 64 x V_WMMA_F32_16X16X4_F32 per wave
__global__ void __launch_bounds__(32)
k_knn(const float* __restrict__ x,
      const float* __restrict__ X, const float* __restrict__ Y, const float* __restrict__ Z,
      const float2* __restrict__ bpk0, const float2* __restrict__ bpk1,
      const float* __restrict__ tf, float* __restrict__ out) {
    __shared__ float tile[256];
    __shared__ float mgv[32 * 5];
    __shared__ int   mgi[32 * 5];
    int lane = threadIdx.x;
    int blk = blockIdx.x;
    int b = blk >> 8;
    int q0 = (blk & 255) * 16;
    int row = lane & 15;       // A-matrix M row == D-tile N column for this lane
    int hf = lane >> 4;
    int q = q0 + row;
    int eidx = b * 4 * Qn + q;               // ego plane index (n = b*4)
    float ex = X[eidx], ey = Y[eidx], ez = Z[eidx];
    float en = ex*ex + ey*ey + ez*ez;
    v2f A; A.x = hf ? ez : ex; A.y = hf ? 1.0f : ey;   // [ex,ey,ez,1] per row

    const float2* bp = hf ? (bpk1 + b * TOPKN) : (bpk0 + b * TOPKN);  // branchless ptr select

    float d0=1e30f,d1=1e30f,d2=1e30f,d3=1e30f,d4=1e30f;
    int   i0=0,i1=0,i2=0,i3=0,i4=0;
    int cbase = hf * 8;

    for (int kt = 0; kt < 64; ++kt) {
        float2 bb = bp[kt * 16 + row];       // one b64 load per lane per tile
        v2f Bv; Bv.x = bb.x; Bv.y = bb.y;
        v8f acc = {};
        acc = __builtin_amdgcn_wmma_f32_16x16x4_f32(
            false, A, false, Bv, (short)0, acc, false, false);  // -2 e.t + |t|^2
#pragma unroll
        for (int i = 0; i < 8; ++i) {
            int r2 = hf ? (8 + i) : i;        // D row M
            tile[r2 * 16 + row] = acc[i];     // column N = row(lane)
        }
        __syncthreads();
        const float4* t4 = (const float4*)(tile + row * 16 + cbase);
        float4 v0 = t4[0], v1 = t4[1];
        int kb0 = kt * 16 + cbase;
        ins5(v0.x + en, kb0 + 0, d0,d1,d2,d3,d4, i0,i1,i2,i3,i4);
        ins5(v0.y + en, kb0 + 1, d0,d1,d2,d3,d4, i0,i1,i2,i3,i4);
        ins5(v0.z + en, kb0 + 2, d0,d1,d2,d3,d4, i0,i1,i2,i3,i4);
        ins5(v0.w + en, kb0 + 3, d0,d1,d2,d3,d4, i0,i1,i2,i3,i4);
        ins5(v1.x + en, kb0 + 4, d0,d1,d2,d3,d4, i0,i1,i2,i3,i4);
        ins5(v1.y + en, kb0 + 5, d0,d1,d2,d3,d4, i0,i1,i2,i3,i4);
        ins5(v1.z + en, kb0 + 6, d0,d1,d2,d3,d4, i0,i1,i2,i3,i4);
        ins5(v1.w + en, kb0 + 7, d0,d1,d2,d3,d4, i0,i1,i2,i3,i4);
        __syncthreads();
    }

    // merge the two half-row lists per row via LDS (all lanes run; lanes>=16 dead)
    mgv[lane*5+0]=d0; mgv[lane*5+1]=d1; mgv[lane*5+2]=d2; mgv[lane*5+3]=d3; mgv[lane*5+4]=d4;
    mgi[lane*5+0]=i0; mgi[lane*5+1]=i1; mgi[lane*5+2]=i2; mgi[lane*5+3]=i3; mgi[lane*5+4]=i4;
    __syncthreads();
    float e0=1e30f,e1=1e30f,e2=1e30f,e3=1e30f,e4=1e30f;
    int   f0=0,f1=0,f2=0,f3=0,f4=0;
#pragma unroll
    for (int j = 0; j < 10; ++j) {
        int src = (j < 5) ? row : (row + 16);
        int slot = (j < 5) ? j : (j - 5);
        ins5(mgv[src*5 + slot], mgi[src*5 + slot], e0,e1,e2,e3,e4, f0,f1,f2,f3,f4);
    }

    // feature max-pool + ego update; all 32 lanes cooperate per row
    const float* tfb = tf + (size_t)(b * TOPKN) * Cn;
    int ch = lane * 8;
    for (int r = 0; r < 16; ++r) {
        int k0 = __shfl(f0, r, 32), k1 = __shfl(f1, r, 32), k2 = __shfl(f2, r, 32),
            k3 = __shfl(f3, r, 32), k4 = __shfl(f4, r, 32);
        const float4* p0 = (const float4*)(tfb + (size_t)k0 * Cn + ch);
        const float4* p1 = (const float4*)(tfb + (size_t)k1 * Cn + ch);
        const float4* p2 = (const float4*)(tfb + (size_t)k2 * Cn + ch);
        const float4* p3 = (const float4*)(tfb + (size_t)k3 * Cn + ch);
        const float4* p4 = (const float4*)(tfb + (size_t)k4 * Cn + ch);
        float4 a0 = p0[0], a1 = p0[1];
        float4 b0 = p1[0], b1 = p1[1];
        a0.x=fmaxf(a0.x,b0.x); a0.y=fmaxf(a0.y,b0.y); a0.z=fmaxf(a0.z,b0.z); a0.w=fmaxf(a0.w,b0.w);
        a1.x=fmaxf(a1.x,b1.x); a1.y=fmaxf(a1.y,b1.y); a1.z=fmaxf(a1.z,b1.z); a1.w=fmaxf(a1.w,b1.w);
        b0 = p2[0]; b1 = p2[1];
        a0.x=fmaxf(a0.x,b0.x); a0.y=fmaxf(a0.y,b0.y); a0.z=fmaxf(a0.z,b0.z); a0.w=fmaxf(a0.w,b0.w);
        a1.x=fmaxf(a1.x,b1.x); a1.y=fmaxf(a1.y,b1.y); a1.z=fmaxf(a1.z,b1.z); a1.w=fmaxf(a1.w,b1.w);
        b0 = p3[0]; b1 = p3[1];
        a0.x=fmaxf(a0.x,b0.x); a0.y=fmaxf(a0.y,b0.y); a0.z=fmaxf(a0.z,b0.z); a0.w=fmaxf(a0.w,b0.w);
        a1.x=fmaxf(a1.x,b1.x); a1.y=fmaxf(a1.y,b1.y); a1.z=fmaxf(a1.z,b1.z); a1.w=fmaxf(a1.w,b1.w);
        b0 = p4[0]; b1 = p4[1];
        a0.x=fmaxf(a0.x,b0.x); a0.y=fmaxf(a0.y,b0.y); a0.z=fmaxf(a0.z,b0.z); a0.w=fmaxf(a0.w,b0.w);
        a1.x=fmaxf(a1.x,b1.x); a1.y=fmaxf(a1.y,b1.y); a1.z=fmaxf(a1.z,b1.z); a1.w=fmaxf(a1.w,b1.w);
        size_t oidx = ((size_t)((q0 + r) * NCH) + b * 4) * Cn + ch;
        const float4* xe = (const float4*)(x + oidx);
        float4 r0 = xe[0], r1 = xe[1];
        r0.x+=a0.x; r0.y+=a0.y; r0.z+=a0.z; r0.w+=a0.w;
        r1.x+=a1.x; r1.y+=a1.y; r1.z+=a1.z; r1.w+=a1.w;
        float4* op = (float4*)(out + oidx);
        op[0] = r0; op[1] = r1;
    }
}

extern "C" void kernel_launch(void* const* d_in, const int* in_sizes, int n_in,
                              void* d_out, int out_size, void* d_ws, size_t ws_size,
                              hipStream_t stream) {
    (void)in_sizes; (void)n_in; (void)out_size; (void)ws_size;
    const float* x   = (const float*)d_in[0];
    const float* ref = (const float*)d_in[1];
    const float* pt  = (const float*)d_in[2];
    const float* hw  = (const float*)d_in[3];
    const float* hb  = (const float*)d_in[4];
    float* out = (float*)d_out;

    float* ws = (float*)d_ws;
    unsigned* keys = (unsigned*)ws;            // B*M        = 98304
    float* Xp = ws + 98304;                    // N*Q        = 131072
    float* Yp = Xp + 131072;
    float* Zp = Yp + 131072;
    int* topidx = (int*)(Zp + 131072);         // B*TOPK     = 8192
    float2* bpk0 = (float2*)(topidx + 8192);   // B*TOPK float2 = 16384 floats
    float2* bpk1 = bpk0 + 8192;                // B*TOPK float2 = 16384 floats
    float* tfp = (float*)(bpk1 + 8192);        // B*TOPK*C   = 2097152

    k_main<<<16384, 256, 0, stream>>>(x, hw, hb, out, keys);
    k_xyz<<<512, 256, 0, stream>>>(ref, pt, Xp, Yp, Zp);
    k_select<<<BATCH, 256, 0, stream>>>(keys, topidx);
    k_gather<<<1024, 256, 0, stream>>>(x, topidx, Xp, Yp, Zp, tfp, bpk0, bpk1);
    k_knn<<<2048, 32, 0, stream>>>(x, Xp, Yp, Zp, bpk0, bpk1, tfp, out);
}